// TopKPool_25890062861050
// MI455X (gfx1250) — compile-verified
//
#include <hip/hip_runtime.h>
#include <hip/hip_bf16.h>

typedef float v2f __attribute__((ext_vector_type(2)));
typedef float v8f __attribute__((ext_vector_type(8)));

#define GB   256        // graphs
#define NN   1024       // nodes per graph
#define CC   64         // channels
#define EE   4194304    // edges
#define KK   512        // kept nodes per graph

// ---------------------------------------------------------------- k_prep
// w_s = w / ||w||, b_s = b / ||w||
__global__ void k_prep(const float* __restrict__ w, const float* __restrict__ b,
                       float* __restrict__ w_s, float* __restrict__ b_s) {
    __shared__ float inv;
    if (threadIdx.x == 0) {
        float ss = 0.f;
        for (int c = 0; c < CC; ++c) { float v = w[c]; ss += v * v; }
        inv = 1.f / sqrtf(ss);
    }
    __syncthreads();
    if (threadIdx.x < CC) w_s[threadIdx.x] = w[threadIdx.x] * inv;
    if (threadIdx.x == 0) b_s[0] = b[0] * inv;
}

// ---------------------------------------------------------------- k_score
// scores = x . w_s + b_s via V_WMMA_F32_16X16X4_F32.
// One wave handles a 16-node tile; K=64 accumulated as 16 chunks of K=4.
// A 16x4 f32 fragment layout (ISA 7.12.2): v0 = K0 (lanes 0-15) / K2 (16-31),
// v1 = K1 / K3, rows M = lane&15.  B 4x16: w replicated across all 16 columns
// -> same per-lane (w[k0], w[k0+1]) pair.  Every column of D equals the score.
__global__ __launch_bounds__(256) void k_score(const float* __restrict__ x,
                                               const float* __restrict__ w_s,
                                               const float* __restrict__ b_s,
                                               float* __restrict__ scores) {
    __shared__ float ws[CC];
    __shared__ float bsh;
    if (threadIdx.x < CC) ws[threadIdx.x] = w_s[threadIdx.x];
    if (threadIdx.x == 0) bsh = b_s[0];
    __syncthreads();

    const int wave = threadIdx.x >> 5;
    const int lane = threadIdx.x & 31;
    const int tile = blockIdx.x * 8 + wave;   // 16 nodes per tile
    const int row0 = tile << 4;
    const int m = lane & 15;                  // row within tile
    const int h = lane >> 4;                  // K-half selector

    const float* xr = x + (size_t)(row0 + m) * CC;

    v8f acc = {};
#pragma unroll
    for (int kc = 0; kc < 16; ++kc) {
        const int k0 = (kc << 2) + (h << 1);
        v2f a;  a.x  = xr[k0]; a.y  = xr[k0 + 1];
        v2f bb; bb.x = ws[k0]; bb.y = ws[k0 + 1];
        acc = __builtin_amdgcn_wmma_f32_16x16x4_f32(
            /*neg_a=*/false, a, /*neg_b=*/false, bb,
            /*c_mod=*/(short)0, acc, /*reuse_a=*/false, /*reuse_b=*/false);
    }

    // D layout: VGPR r holds M=r (lanes 0-15) / M=r+8 (lanes 16-31); all N equal.
    if (m == 0) {
        const float bias = bsh;
#pragma unroll
        for (int r = 0; r < 8; ++r)
            scores[row0 + (h << 3) + r] = acc[r] + bias;
    }
}

// ---------------------------------------------------------------- k_topk
// One block per graph: bitonic sort 1024 (score desc, idx asc), keep 512,
// build perm / batch_p outputs and the new_idx relabel table.
__global__ __launch_bounds__(1024) void k_topk(const float* __restrict__ scores,
                                               int* __restrict__ perm_i,
                                               int* __restrict__ new_idx,
                                               float* __restrict__ out_perm,
                                               float* __restrict__ out_batchp) {
    __shared__ float ss[NN];
    __shared__ int   si[NN];
    const int g = blockIdx.x;
    const int t = threadIdx.x;

    ss[t] = scores[g * NN + t];
    si[t] = t;
    __syncthreads();

    for (int k = 2; k <= NN; k <<= 1) {
        for (int j = k >> 1; j > 0; j >>= 1) {
            const int ixj = t ^ j;
            if (ixj > t) {
                const float s1 = ss[t], s2 = ss[ixj];
                const int   i1 = si[t], i2 = si[ixj];
                const bool up    = ((t & k) == 0);  // descending segment
                const bool worse = (s1 < s2) || (s1 == s2 && i1 > i2);
                if (up == worse) {
                    ss[t] = s2; ss[ixj] = s1;
                    si[t] = i2; si[ixj] = i1;
                }
            }
            __syncthreads();
        }
    }

    new_idx[g * NN + t] = -1;
    __syncthreads();

    if (t < KK) {
        const int li = si[t];             // local node id kept at rank t
        const int gi = g * NN + li;       // global old id
        const int pi = g * KK + t;        // global new id
        perm_i[pi]     = gi;
        new_idx[gi]    = pi;
        out_perm[pi]   = (float)gi;
        out_batchp[pi] = (float)g;
    }
}

// ---------------------------------------------------------------- k_gather
// x_p[row] = x[perm[row]]  (float4 vectorized row copy)
__global__ __launch_bounds__(256) void k_gather(const float* __restrict__ x,
                                                const int* __restrict__ perm_i,
                                                float* __restrict__ xp) {
    const int i   = blockIdx.x * 256 + threadIdx.x;  // over B*K*16 float4s
    const int row = i >> 4;
    const int q   = i & 15;
    const int src = perm_i[row];
    const float4 v = ((const float4*)(x + (size_t)src * CC))[q];
    ((float4*)(xp + (size_t)row * CC))[q] = v;
}

// ---------------------------------------------------------------- k_edges
// Relabel endpoints through new_idx, mask dropped edges.
__global__ __launch_bounds__(256) void k_edges(const int* __restrict__ ei,
                                               const float* __restrict__ ea,
                                               const int* __restrict__ new_idx,
                                               float* __restrict__ out_ei,
                                               float* __restrict__ out_ea,
                                               float* __restrict__ out_mask) {
    const int e = blockIdx.x * 256 + threadIdx.x;
    const int r = new_idx[ei[e]];
    const int c = new_idx[ei[e + EE]];
    const bool mk = (r >= 0) && (c >= 0);
    out_ei[e]      = (float)(mk ? r : -1);
    out_ei[e + EE] = (float)(mk ? c : -1);
    out_ea[e]      = mk ? ea[e] : 0.f;
    out_mask[e]    = mk ? 1.f : 0.f;
}

// ---------------------------------------------------------------- launch
extern "C" void kernel_launch(void* const* d_in, const int* in_sizes, int n_in,
                              void* d_out, int out_size, void* d_ws, size_t ws_size,
                              hipStream_t stream) {
    const float* x  = (const float*)d_in[0];
    const float* w  = (const float*)d_in[1];
    const float* b  = (const float*)d_in[2];
    const int*   ei = (const int*)  d_in[3];
    const float* ea = (const float*)d_in[4];
    // d_in[5] (batch) not needed: batch_p[g*K+t] == g for equal-size graphs.

    // Output layout (floats), reference return order.
    float* out  = (float*)d_out;
    float* o_xp = out;                       // B*K*C = 8388608
    float* o_ei = o_xp + (size_t)GB * KK * CC;   // 2*E  = 8388608
    float* o_ea = o_ei + 2 * (size_t)EE;         // E    = 4194304
    float* o_bp = o_ea + (size_t)EE;             // B*K  = 131072
    float* o_sc = o_bp + (size_t)GB * KK;        // B*N  = 262144
    float* o_pm = o_sc + (size_t)GB * NN;        // B*K  = 131072
    float* o_mk = o_pm + (size_t)GB * KK;        // E    = 4194304

    // Workspace
    float* w_s     = (float*)d_ws;                                   // 64
    float* b_s     = w_s + CC;                                       // 1
    int*   perm_i  = (int*)((char*)d_ws + 512);                      // B*K ints
    int*   new_idx = (int*)((char*)d_ws + 512 + (size_t)GB * KK * 4);// B*N ints

    k_prep  <<<1, 64, 0, stream>>>(w, b, w_s, b_s);
    k_score <<<(GB * NN) / (16 * 8), 256, 0, stream>>>(x, w_s, b_s, o_sc);
    k_topk  <<<GB, NN, 0, stream>>>(o_sc, perm_i, new_idx, o_pm, o_bp);
    k_gather<<<(GB * KK * 16) / 256, 256, 0, stream>>>(x, perm_i, o_xp);
    k_edges <<<EE / 256, 256, 0, stream>>>(ei, ea, new_idx, o_ei, o_ea, o_mk);
}